// SimpleGCN_54786602828183
// MI455X (gfx1250) — compile-verified
//
#include <hip/hip_runtime.h>
#include <math.h>

typedef __attribute__((ext_vector_type(2))) float v2f;
typedef __attribute__((ext_vector_type(8))) float v8f;

// ---------------------------------------------------------------------------
// Degree / normalization kernels
// ---------------------------------------------------------------------------
__global__ void k_deg_init(float* __restrict__ deg, int n) {
    int i = blockIdx.x * blockDim.x + threadIdx.x;
    if (i < n) deg[i] = 1.0f;  // self-loop contributes 1 to every node's degree
}

__global__ void k_deg_accum(const long long* __restrict__ ei, float* __restrict__ deg, int E) {
    int e = blockIdx.x * blockDim.x + threadIdx.x;
    if (e < E) {
        int d = (int)ei[(size_t)E + e];  // edge_index[1][e] = dst
        atomicAdd(&deg[d], 1.0f);
    }
}

__global__ void k_dinv(float* __restrict__ deg, int n) {
    int i = blockIdx.x * blockDim.x + threadIdx.x;
    if (i < n) {
        float v = deg[i];
        deg[i] = (v > 0.0f) ? rsqrtf(v) : 0.0f;  // in-place: deg -> d^{-1/2}
    }
}

// ---------------------------------------------------------------------------
// Dense transform H = A(MxK) * W(KxN) via V_WMMA_F32_16X16X4_F32 (fp32 exact).
// One wave32 computes a 16-row output tile with N/16 f32 accumulators.
// VGPR layouts per CDNA5 ISA 7.12.2:
//   A 16x4 f32 : lanes 0-15 -> M=lane, {v0,v1}={K+0,K+1}; lanes 16-31 -> {K+2,K+3}
//   B 4x16 f32 : lanes 0-15 -> N=lane, {v0,v1}={K+0,K+1}; lanes 16-31 -> {K+2,K+3}
//   C/D 16x16  : VGPR r -> M=r (lanes 0-15) / M=r+8 (lanes 16-31), N=lane&15
// ---------------------------------------------------------------------------
template <int K, int N>
__global__ void k_gemm_wmma(const float* __restrict__ A, const float* __restrict__ W,
                            float* __restrict__ H, int M) {
    const int lane = threadIdx.x & 31;
    const int wave = blockIdx.x * (blockDim.x >> 5) + (threadIdx.x >> 5);
    const int row0 = wave * 16;
    if (row0 >= M) return;

    if (row0 + 16 > M) {
        // Wave-uniform scalar tail (not reached when M % 16 == 0).
        for (int r = row0; r < M; ++r)
            for (int c = lane; c < N; c += 32) {
                float s = 0.0f;
                for (int k = 0; k < K; ++k) s += A[(size_t)r * K + k] * W[k * N + c];
                H[(size_t)r * N + c] = s;
            }
        return;
    }

    const int half = lane >> 4;   // 0: lanes 0-15, 1: lanes 16-31
    const int l15  = lane & 15;
    constexpr int NT = N / 16;

    v8f acc[NT];
#pragma unroll
    for (int t = 0; t < NT; ++t) acc[t] = {};

    const float* arow = A + (size_t)(row0 + l15) * K;

#pragma unroll
    for (int k = 0; k < K; k += 4) {
        v2f a;
        a.x = arow[k + 2 * half + 0];
        a.y = arow[k + 2 * half + 1];
#pragma unroll
        for (int t = 0; t < NT; ++t) {
            v2f b;
            const int n = t * 16 + l15;
            b.x = W[(k + 2 * half + 0) * N + n];
            b.y = W[(k + 2 * half + 1) * N + n];
            acc[t] = __builtin_amdgcn_wmma_f32_16x16x4_f32(
                false, a, false, b, (short)0, acc[t], false, false);
        }
    }

#pragma unroll
    for (int r = 0; r < 8; ++r) {
        const int row = row0 + r + half * 8;
#pragma unroll
        for (int t = 0; t < NT; ++t)
            H[(size_t)row * N + t * 16 + l15] = acc[t][r];
    }
}

// ---------------------------------------------------------------------------
// AGG[i,:] = H[i,:] * dinv[i]^2   (self-loop term; also initializes the buffer)
// ---------------------------------------------------------------------------
template <int F>
__global__ void k_self_init(const float* __restrict__ H, const float* __restrict__ dinv,
                            float* __restrict__ AGG, int n) {
    long long idx = (long long)blockIdx.x * blockDim.x + threadIdx.x;
    if (idx < (long long)n * F) {
        int i = (int)(idx / F);
        float di = dinv[i];
        AGG[idx] = H[idx] * di * di;
    }
}

// ---------------------------------------------------------------------------
// Edge scatter: AGG[dst,:] += H[src,:] * dinv[src]*dinv[dst]
// One thread per (edge, 4-feature chunk); float4 gather + 4 f32 atomics (L2).
// ---------------------------------------------------------------------------
template <int F>
__global__ void k_scatter(const long long* __restrict__ ei, const float* __restrict__ H,
                          const float* __restrict__ dinv, float* __restrict__ AGG, int E) {
    constexpr int QP = F / 4;
    long long t = (long long)blockIdx.x * blockDim.x + threadIdx.x;
    if (t >= (long long)E * QP) return;
    int e = (int)(t / QP);
    int q = (int)(t % QP);
    int s = (int)ei[e];
    int d = (int)ei[(size_t)E + e];
    float norm = dinv[s] * dinv[d];
    float4 v = *(const float4*)(H + (size_t)s * F + q * 4);
    float* out = AGG + (size_t)d * F + q * 4;
    atomicAdd(out + 0, v.x * norm);
    atomicAdd(out + 1, v.y * norm);
    atomicAdd(out + 2, v.z * norm);
    atomicAdd(out + 3, v.w * norm);
}

// ---------------------------------------------------------------------------
// Epilogues
// ---------------------------------------------------------------------------
template <int F>
__global__ void k_bias_relu(const float* __restrict__ AGG, const float* __restrict__ b,
                            float* __restrict__ OUT, int n) {
    long long idx = (long long)blockIdx.x * blockDim.x + threadIdx.x;
    if (idx < (long long)n * F) {
        int f = (int)(idx % F);
        OUT[idx] = fmaxf(AGG[idx] + b[f], 0.0f);
    }
}

template <int F>
__global__ void k_bias_sigmoid(float* __restrict__ OUT, const float* __restrict__ b, int n) {
    long long idx = (long long)blockIdx.x * blockDim.x + threadIdx.x;
    if (idx < (long long)n * F) {
        int f = (int)(idx % F);
        float v = OUT[idx] + b[f];
        OUT[idx] = 1.0f / (1.0f + expf(-v));
    }
}

// ---------------------------------------------------------------------------
extern "C" void kernel_launch(void* const* d_in, const int* in_sizes, int n_in,
                              void* d_out, int out_size, void* d_ws, size_t ws_size,
                              hipStream_t stream) {
    const float*     x  = (const float*)d_in[0];
    const long long* ei = (const long long*)d_in[1];  // int64 edge_index, shape (2,E)
    const float*     W1 = (const float*)d_in[2];
    const float*     b1 = (const float*)d_in[3];
    const float*     W2 = (const float*)d_in[4];
    const float*     b2 = (const float*)d_in[5];
    float* out = (float*)d_out;

    const int n = in_sizes[0] / 64;   // 100000 nodes
    const int E = in_sizes[1] / 2;    // 3.2M edges

    // Workspace layout (256B aligned): dinv | h1 | agg1 | h2
    char* ws = (char*)d_ws;
    size_t off = 0;
    auto alloc = [&](size_t bytes) { void* p = ws + off; off += (bytes + 255) & ~(size_t)255; return p; };
    float* dinv = (float*)alloc((size_t)n * sizeof(float));
    float* h1   = (float*)alloc((size_t)n * 32 * sizeof(float));
    float* agg1 = (float*)alloc((size_t)n * 32 * sizeof(float));
    float* h2   = (float*)alloc((size_t)n * 64 * sizeof(float));
    (void)ws_size; (void)n_in; (void)out_size;

    const int B = 256;
    // --- normalization ---
    k_deg_init <<<(n + B - 1) / B, B, 0, stream>>>(dinv, n);
    k_deg_accum<<<(E + B - 1) / B, B, 0, stream>>>(ei, dinv, E);
    k_dinv     <<<(n + B - 1) / B, B, 0, stream>>>(dinv, n);

    const int waves = (n + 15) / 16;
    const int gemmBlocks = (waves + 7) / 8;  // 8 wave32 per 256-thread block

    // --- layer 1: F = 32 ---
    k_gemm_wmma<64, 32><<<gemmBlocks, B, 0, stream>>>(x, W1, h1, n);
    long long t1 = (long long)n * 32;
    k_self_init<32><<<(int)((t1 + B - 1) / B), B, 0, stream>>>(h1, dinv, agg1, n);
    long long s1 = (long long)E * 8;
    k_scatter<32><<<(int)((s1 + B - 1) / B), B, 0, stream>>>(ei, h1, dinv, agg1, E);
    k_bias_relu<32><<<(int)((t1 + B - 1) / B), B, 0, stream>>>(agg1, b1, h1, n);  // a1 -> h1 buf

    // --- layer 2: F = 64, accumulate directly into d_out ---
    k_gemm_wmma<32, 64><<<gemmBlocks, B, 0, stream>>>(h1, W2, h2, n);
    long long t2 = (long long)n * 64;
    k_self_init<64><<<(int)((t2 + B - 1) / B), B, 0, stream>>>(h2, dinv, out, n);
    long long s2 = (long long)E * 16;
    k_scatter<64><<<(int)((s2 + B - 1) / B), B, 0, stream>>>(ei, h2, dinv, out, E);
    k_bias_sigmoid<64><<<(int)((t2 + B - 1) / B), B, 0, stream>>>(out, b2, n);
}